// DropBlock_14516989460837
// MI455X (gfx1250) — compile-verified
//
#include <hip/hip_runtime.h>
#include <stdint.h>

// Problem constants (B=64, C=640, H=W=32, block_size=5 -> mask is 28x28)
#define NELEM    41943040u          // 64*640*32*32
#define NPLANES  40960              // 64*640
#define MAREA    784                // 28*28
#define COUNT_M  41943040.0f

#define SMP_PITCH  36               // 28 cols + 4 zero pad each side
#define SROW_PITCH 33               // 32 cols + 1 to break bank aliasing
#define SROW_ROWS  36               // 28 rows + 4 zero pad rows each side

// ---- CDNA5 async global->LDS path (ASYNCcnt), with safe fallbacks ----------
#if defined(__HIP_DEVICE_COMPILE__) && __has_builtin(__builtin_amdgcn_global_load_async_to_lds_b128)
#define USE_ASYNC_LDS 1
#else
#define USE_ASYNC_LDS 0
#endif

// Pointee type the async-LDS builtins expect: int __vector(4) (16 bytes).
typedef int v4i_b128 __attribute__((vector_size(16)));

#if defined(__HIP_DEVICE_COMPILE__)
#if __has_builtin(__builtin_amdgcn_s_wait_asynccnt)
#define WAIT_ASYNC() __builtin_amdgcn_s_wait_asynccnt(0)
#else
#define WAIT_ASYNC() asm volatile("s_wait_asynccnt 0" ::: "memory")
#endif
#else
#define WAIT_ASYNC()
#endif

// Stage one 28x28 mask plane into zero-padded LDS, then separable 5x5
// dilation (window [o-4, o], zero outside [0,27]) with fixed-trip 5-tap
// max loops.  Each thread ends with expanded values for output pixels
// 4t..4t+3 (h = t>>3, w0 = (t&7)*4) of the 32x32 plane.
__device__ __forceinline__ void dilate_plane(const float* __restrict__ mplane,
                                             float* smp,    // [28][36]
                                             float* srowp,  // [36][33]
                                             float& e0, float& e1,
                                             float& e2, float& e3) {
  const int t = threadIdx.x;

  // Zero smp pad columns: rows 0..27, cols 0..3 and 32..35 (224 entries).
  if (t < 224) {
    const int r = t >> 3;                    // 0..27
    const int c = t & 7;                     // 0..7
    smp[r * SMP_PITCH + (c < 4 ? c : c + 28)] = 0.0f;
  }
  // Zero srowp pad rows: rows 0..3 and 32..35, cols 0..31 (256 entries).
  {
    const int r = t >> 5;                    // 0..7
    const int c = t & 31;
    const int row = (r < 4) ? r : r + 28;    // 0..3 or 32..35
    srowp[row * SROW_PITCH + c] = 0.0f;
  }

  // Stage 0: 28 rows x 7 b128 async copies into smp[i][4..31].
  // Row base = i*144+16 bytes -> 16B aligned; 28 elems = 7 x 4 contiguous.
#if USE_ASYNC_LDS
  if (t < 196) {
    const int i = t / 7;                     // row 0..27
    const int c = t - i * 7;                 // chunk 0..6
    __builtin_amdgcn_global_load_async_to_lds_b128(
        (v4i_b128*)(mplane + i * 28 + c * 4),
        (v4i_b128*)(smp + i * SMP_PITCH + 4 + c * 4),
        0, 0);
  }
  WAIT_ASYNC();
#else
  for (int i = t; i < MAREA; i += 256) {
    const int r = i / 28;
    const int c = i - r * 28;
    smp[r * SMP_PITCH + 4 + c] = mplane[i];
  }
#endif
  __syncthreads();

  // Stage 1: row dilation, fixed 5-tap -> srowp[i+4][w], i=0..27, w=0..31.
  for (int idx = t; idx < 28 * 32; idx += 256) {
    const int i = idx >> 5;
    const int w = idx & 31;
    const float* p = smp + i * SMP_PITCH + w;   // padded: cols w..w+4
    const float m =
        fmaxf(fmaxf(fmaxf(p[0], p[1]), fmaxf(p[2], p[3])), p[4]);
    srowp[(i + 4) * SROW_PITCH + w] = m;
  }
  __syncthreads();

  // Stage 2: column dilation, fixed 5-tap over padded rows h..h+4.
  const int h  = t >> 3;
  const int w0 = (t & 7) << 2;
  const float* p0 = srowp + h * SROW_PITCH + w0;
  e0 = p0[0]; e1 = p0[1]; e2 = p0[2]; e3 = p0[3];
#pragma unroll
  for (int r = 1; r < 5; ++r) {
    const float* p = srowp + (h + r) * SROW_PITCH + w0;
    e0 = fmaxf(e0, p[0]);
    e1 = fmaxf(e1, p[1]);
    e2 = fmaxf(e2, p[2]);
    e3 = fmaxf(e3, p[3]);
  }
}

__global__ void dropblock_zero(unsigned* counter) { *counter = 0u; }

// Pass 1: per-plane dilation, bit-pack expanded mask, count dropped pixels.
__global__ __launch_bounds__(256) void dropblock_pass1(
    const float* __restrict__ mask, unsigned* __restrict__ counter,
    unsigned* __restrict__ bits, int store_bits) {
  __shared__ __align__(16) float smp[28 * SMP_PITCH];
  __shared__ float srowp[SROW_ROWS * SROW_PITCH];
  __shared__ unsigned snib[256];
  __shared__ unsigned scount;

  const int t = threadIdx.x;
  const int plane = blockIdx.x;
  if (t == 0) scount = 0u;

  float e0, e1, e2, e3;
  dilate_plane(mask + (size_t)plane * MAREA, smp, srowp, e0, e1, e2, e3);

  const unsigned nib = (unsigned)(e0 > 0.0f) | ((unsigned)(e1 > 0.0f) << 1) |
                       ((unsigned)(e2 > 0.0f) << 2) |
                       ((unsigned)(e3 > 0.0f) << 3);
  snib[t] = nib;
  __syncthreads();

  if (t < 32) {  // row t: combine 8 nibbles -> one 32-bit row word
    unsigned word = 0u;
#pragma unroll
    for (int k = 0; k < 8; ++k) word |= snib[t * 8 + k] << (4 * k);
    if (store_bits) bits[(size_t)plane * 32 + t] = word;
    atomicAdd(&scount, (unsigned)__popc(word));
  }
  __syncthreads();
  if (t == 0) atomicAdd(counter, scount);
}

// Pass 2 (fast path): flat float4 scaling using the packed bitmask.
__global__ __launch_bounds__(256) void dropblock_pass2_bits(
    const float* __restrict__ x, float* __restrict__ out,
    const unsigned* __restrict__ bits, const unsigned* __restrict__ counter) {
  const unsigned cnt = *counter;                       // E = #expanded ones
  const float scale = COUNT_M / (COUNT_M - (float)cnt);
  const size_t gid = (size_t)blockIdx.x * blockDim.x + threadIdx.x;
  const size_t base = gid * 4;
  if (base >= (size_t)NELEM) return;
  const size_t plane = base >> 10;
  const unsigned within = (unsigned)(base & 1023u);
  const unsigned h = within >> 5, w0 = within & 31u;   // w0 multiple of 4
  const unsigned word = bits[plane * 32 + h];
  const float4 v = ((const float4*)x)[gid];
  float4 o;
  o.x = ((word >> (w0 + 0)) & 1u) ? 0.0f : v.x * scale;
  o.y = ((word >> (w0 + 1)) & 1u) ? 0.0f : v.y * scale;
  o.z = ((word >> (w0 + 2)) & 1u) ? 0.0f : v.z * scale;
  o.w = ((word >> (w0 + 3)) & 1u) ? 0.0f : v.w * scale;
  ((float4*)out)[gid] = o;
}

// Pass 2 (fallback if ws too small for the bitmask): recompute dilation.
__global__ __launch_bounds__(256) void dropblock_pass2_recompute(
    const float* __restrict__ x, const float* __restrict__ mask,
    float* __restrict__ out, const unsigned* __restrict__ counter) {
  __shared__ __align__(16) float smp[28 * SMP_PITCH];
  __shared__ float srowp[SROW_ROWS * SROW_PITCH];

  const int t = threadIdx.x;
  const int plane = blockIdx.x;
  float e0, e1, e2, e3;
  dilate_plane(mask + (size_t)plane * MAREA, smp, srowp, e0, e1, e2, e3);

  const unsigned cnt = *counter;
  const float scale = COUNT_M / (COUNT_M - (float)cnt);
  const size_t vbase = (size_t)plane * 256 + t;        // float4 index
  const float4 v = ((const float4*)x)[vbase];
  float4 o;
  o.x = (e0 > 0.0f) ? 0.0f : v.x * scale;
  o.y = (e1 > 0.0f) ? 0.0f : v.y * scale;
  o.z = (e2 > 0.0f) ? 0.0f : v.z * scale;
  o.w = (e3 > 0.0f) ? 0.0f : v.w * scale;
  ((float4*)out)[vbase] = o;
}

extern "C" void kernel_launch(void* const* d_in, const int* in_sizes, int n_in,
                              void* d_out, int out_size, void* d_ws, size_t ws_size,
                              hipStream_t stream) {
  (void)in_sizes; (void)n_in; (void)out_size;
  const float* x    = (const float*)d_in[0];   // [64,640,32,32] f32
  const float* mask = (const float*)d_in[1];   // [64,640,28,28] f32
  float* out = (float*)d_out;

  unsigned* counter = (unsigned*)d_ws;                       // 4B @ offset 0
  unsigned* bits    = (unsigned*)((char*)d_ws + 16);         // 32 words/plane
  const size_t need = 16 + (size_t)NPLANES * 32 * sizeof(unsigned);
  const bool use_bits = ws_size >= need;

  dropblock_zero<<<1, 1, 0, stream>>>(counter);
  dropblock_pass1<<<NPLANES, 256, 0, stream>>>(mask, counter, bits,
                                               use_bits ? 1 : 0);
  if (use_bits) {
    dropblock_pass2_bits<<<NELEM / 4 / 256, 256, 0, stream>>>(x, out, bits,
                                                              counter);
  } else {
    dropblock_pass2_recompute<<<NPLANES, 256, 0, stream>>>(x, mask, out,
                                                           counter);
  }
}